// OneNet_67164698575367
// MI455X (gfx1250) — compile-verified
//
#include <hip/hip_runtime.h>
#include <math.h>

// CDNA5 / gfx1250, wave32.
// D = A(16xK) x B(Kx16) + C(16x16), K=4 per V_WMMA_F32_16X16X4_F32.
typedef __attribute__((ext_vector_type(2))) float v2f;
typedef __attribute__((ext_vector_type(8))) float v8f;

#define LOG2E 1.4426950408889634f
#define LN2 0.6931471805599453f

__device__ __forceinline__ v8f wmma4(v2f a, v2f b, v8f c) {
  // 8 args: (neg_a, A, neg_b, B, c_mod, C, reuse_a, reuse_b)
  return __builtin_amdgcn_wmma_f32_16x16x4_f32(false, a, false, b, (short)0, c,
                                               false, false);
}

// Raw v_exp_f32 / v_log_f32 / v_rcp_f32 (no libm denorm scaffolding; our
// softplus exp2 args are <= 0 and log2 args are >= 1, so it is safe).
__device__ __forceinline__ float exp2_raw(float x) {
  return __builtin_amdgcn_exp2f(x);
}
__device__ __forceinline__ float log2_raw(float x) {
  return __builtin_amdgcn_logf(x);
}
__device__ __forceinline__ float rcp_raw(float x) {
  return __builtin_amdgcn_rcpf(x);
}

// softplus(x) = max(x,0) + ln2 * log2(1 + 2^(-|x|*log2e))
__device__ __forceinline__ float softplus_f(float x) {
  const float e = exp2_raw(-fabsf(x) * LOG2E);
  return fmaxf(x, 0.f) + LN2 * log2_raw(1.f + e);
}

// Half-wave swap (lane <-> lane^16) via DS_SWIZZLE_B32, group-of-32 mode:
// and_mask=0x1f, or_mask=0, xor_mask=0x10  -> imm 0x401f (SWAPX16).
__device__ __forceinline__ float swap16(float x) {
  return __int_as_float(__builtin_amdgcn_ds_swizzle(__float_as_int(x), 0x401f));
}

__global__ __launch_bounds__(128) void ecm_scan_wmma(
    const float* __restrict__ X,   // (B, L, 5): t, I, T, dummy, SOC
    const float* __restrict__ SC,  // (B, 3)
    const float* __restrict__ pW1, // (5, 10)
    const float* __restrict__ pb1, // (10)
    const float* __restrict__ pW2, // (10, 5)
    const float* __restrict__ pb2, // (5)
    const float* __restrict__ rW1, // (5, 1)
    const float* __restrict__ rb1, // (1)
    const float* __restrict__ rW2, // (1, 1)
    const float* __restrict__ rb2, // (1)
    float* __restrict__ out,       // (B, 1)
    int Bn, int Ln) {
  const int lane = threadIdx.x & 31;
  const int wave = threadIdx.x >> 5;
  const int half = lane >> 4;  // 0: lanes 0-15, 1: lanes 16-31
  const int n    = lane & 15;  // column / batch-in-group index
  const int grp  = blockIdx.x * (blockDim.x >> 5) + wave;
  if (grp * 16 >= Bn) return;  // wave-uniform exit (keeps EXEC all-1 for WMMA)

  const int b  = grp * 16 + n;          // this lane's batch row
  const int bl = (b < Bn) ? b : Bn - 1; // clamped for loads

  // ---- time-constant per-batch inputs -------------------------------------
  const float sc0 = SC[bl * 3 + 0];
  const float sc1 = SC[bl * 3 + 1];
  const float sc2 = SC[bl * 3 + 2];

  // ---- fold weights into WMMA operand registers ---------------------------
  // Layer 1: A1 = W1^T (10x5) zero-padded to 16x8.  A-layout: lane holds row
  // M=n; VGPR pair holds K = {2*half, 2*half+1} (chunk 0) / {4+2*half, ...}.
  v2f a1_lo, a1_hi;
  a1_lo.x = (n < 10) ? pW1[(2 * half + 0) * 10 + n] : 0.f;
  a1_lo.y = (n < 10) ? pW1[(2 * half + 1) * 10 + n] : 0.f;
  a1_hi.x = (half == 0 && n < 10) ? pW1[4 * 10 + n] : 0.f;  // K=4; K=5..7 pad
  a1_hi.y = 0.f;
  // Layer 2: A2 = W2^T (5x10) zero-padded to 16x12 (three K=4 chunks).
  // Pre-scaled by -0.01*log2e so the WMMA output feeds exp2 directly:
  //   sigmoid(0.01*p) == rcp(1 + exp2(p_scaled)).
  const float A2S = -0.01f * LOG2E;
  v2f a2_1, a2_2, a2_3;
  a2_1.x = (n < 5) ? A2S * pW2[(2 * half + 0) * 5 + n] : 0.f;
  a2_1.y = (n < 5) ? A2S * pW2[(2 * half + 1) * 5 + n] : 0.f;
  a2_2.x = (n < 5) ? A2S * pW2[(4 + 2 * half) * 5 + n] : 0.f;
  a2_2.y = (n < 5) ? A2S * pW2[(5 + 2 * half) * 5 + n] : 0.f;
  a2_3.x = (half == 0 && n < 5) ? A2S * pW2[8 * 5 + n] : 0.f;  // K=8,9
  a2_3.y = (half == 0 && n < 5) ? A2S * pW2[9 * 5 + n] : 0.f;  // K=10,11 pad
  // Biases ride as the initial C accumulator: C[m][:] = bias[m].
  // C-layout: VGPR r holds row m = r + 8*half.
  v8f cb1, cb2;
#pragma unroll
  for (int r = 0; r < 8; ++r) {
    const int m = r + 8 * half;
    cb1[r] = (m < 10) ? pb1[m] : 0.f;
    cb2[r] = (m < 5) ? A2S * pb2[m] : 0.f;
  }
  // r-network scalars (used only at l==0)
  const float rw10 = rW1[0], rw11 = rW1[1], rw12 = rW1[2], rw13 = rW1[3],
              rw14 = rW1[4];
  const float rb1v = rb1[0], rw2v = rW2[0], rb2v = rb2[0];

  // theta bounds (compile-time constants of the reference)
  const float LB0 = 0.005f, LB1 = 0.025f, LB2 = 0.1f, LB3 = 0.0f, LB4 = 0.002f;
  const float D0 = 0.02f - LB0, D1 = 0.07f - LB1, D2 = 1.0f - LB2,
              D3 = 0.055f - LB3, D4 = 0.025f - LB4;

  const float* xp = X + (size_t)bl * (size_t)Ln * 5;

  float UH = 0.f, U1v = 0.f;      // scan state y = (U_H, U_1)
  float fUH = 0.f, fU1 = 0.f;     // pending derivative from step l-1
  float t_prev = 0.f;
  float sum = 0.f, cnt = 0.f;     // nan-mean accumulators

  for (int l = 0; l < Ln; ++l) {
    // ---- per-step loads: only lanes 0-15 stream their batch row ----------
    float t_l = 0.f, I_l = 0.f, T_l = 0.f, S_l = 0.f;
    if (half == 0) {
      const float* p = xp + (size_t)l * 5;
      t_l = p[0];
      I_l = p[1];
      T_l = p[2];
      S_l = p[4];
    }

    // ---- layer 1: H^T = W1^T @ X^T + b1  (x = [SOC, T, sc0, sc1, sc2]) ---
    // B-layout (KxN): lanes 0-15 carry K={0,1} (SOC,T), lanes 16-31 K={2,3}
    // (sc0,sc1); second chunk: lanes 0-15 K=4 (sc2), rest zero padding.
    v2f bx1, bx2;
    bx1.x = half ? sc0 : S_l;
    bx1.y = half ? sc1 : T_l;
    bx2.x = half ? 0.f : sc2;
    bx2.y = 0.f;
    v8f h = cb1;
    h = wmma4(a1_lo, bx1, h);
    h = wmma4(a1_hi, bx2, h);
#pragma unroll
    for (int r = 0; r < 8; ++r) h[r] = softplus_f(h[r]);

    // ---- C-layout -> B-layout for layer 2: swap wave halves --------------
    const float s0 = swap16(h[0]);  // rows 8,9 for lanes 0-15
    const float s1 = swap16(h[1]);
    const float s2 = swap16(h[2]);  // rows 2,3 for lanes 16-31
    const float s3 = swap16(h[3]);
    const float s6 = swap16(h[6]);  // rows 6,7 for lanes 16-31
    const float s7 = swap16(h[7]);
    v2f bh1, bh2, bh3;
    bh1.x = half ? s2 : h[0];
    bh1.y = half ? s3 : h[1];
    bh2.x = half ? s6 : h[4];
    bh2.y = half ? s7 : h[5];
    bh3.x = half ? 0.f : s0;  // K=10,11 are zero padding
    bh3.y = half ? 0.f : s1;

    // ---- layer 2: P'^T = (-0.01*log2e) * (W2^T @ H^T + b2) ---------------
    v8f pv = cb2;
    pv = wmma4(a2_1, bh1, pv);
    pv = wmma4(a2_2, bh2, pv);
    pv = wmma4(a2_3, bh3, pv);

    // lanes 0-15, rows 0..4 hold scaled p_k; sigmoid = rcp(1 + 2^p')
    const float R1 = LB0 + D0 * rcp_raw(1.f + exp2_raw(pv[0]));
    const float RC = LB1 + D1 * rcp_raw(1.f + exp2_raw(pv[1]));
    const float OC = LB2 + D2 * rcp_raw(1.f + exp2_raw(pv[2]));
    const float MH = LB3 + D3 * rcp_raw(1.f + exp2_raw(pv[3]));
    const float KH = LB4 + D4 * rcp_raw(1.f + exp2_raw(pv[4]));

    // ---- scan (l uniform across the wave) --------------------------------
    if (l == 0) {
      const float z =
          rb1v + S_l * rw10 + T_l * rw11 + sc0 * rw12 + sc1 * rw13 + sc2 * rw14;
      const float rr = softplus_f(z) * rw2v + rb2v;
      const float Rs = sc2 * (1.f + rr);
      UH = 0.f;
      U1v = -OC - I_l * Rs;
    } else {
      const float dt = t_l - t_prev;
      UH += dt * fUH;
      U1v += dt * fU1;
    }
    const float v = (-OC - UH - U1v) * rcp_raw(I_l);
    if (v == v) {  // nanmean: skip NaN
      sum += v;
      cnt += 1.f;
    }
    fUH = -KH * I_l * (-MH - UH);
    fU1 = RC * (R1 * I_l - U1v);
    t_prev = t_l;
  }

  if (half == 0 && b < Bn) out[b] = sum * rcp_raw(cnt);
}

extern "C" void kernel_launch(void* const* d_in, const int* in_sizes, int n_in,
                              void* d_out, int out_size, void* d_ws,
                              size_t ws_size, hipStream_t stream) {
  const float* X   = (const float*)d_in[0];
  const float* SC  = (const float*)d_in[1];
  const float* pW1 = (const float*)d_in[2];
  const float* pb1 = (const float*)d_in[3];
  const float* pW2 = (const float*)d_in[4];
  const float* pb2 = (const float*)d_in[5];
  const float* rW1 = (const float*)d_in[6];
  const float* rb1 = (const float*)d_in[7];
  const float* rW2 = (const float*)d_in[8];
  const float* rb2 = (const float*)d_in[9];
  float* out = (float*)d_out;

  const int Bn = in_sizes[1] / 3;              // SC is (B, 3)
  const int Ln = in_sizes[0] / (Bn * 5);       // X is (B, L, 5)
  const int groups = (Bn + 15) / 16;           // 16 batch rows per wave
  const int wavesPerBlock = 4;                 // 128 threads
  const int blocks = (groups + wavesPerBlock - 1) / wavesPerBlock;

  ecm_scan_wmma<<<blocks, wavesPerBlock * 32, 0, stream>>>(
      X, SC, pW1, pb1, pW2, pb2, rW1, rb1, rW2, rb2, out, Bn, Ln);
}